// alad_metad_53815940218958
// MI455X (gfx1250) — compile-verified
//
#include <hip/hip_runtime.h>

typedef __attribute__((ext_vector_type(2))) float v2f;
typedef __attribute__((ext_vector_type(8))) float v8f;

#define DEGF 57.29577951308232f /* 180/pi */

__device__ __forceinline__ void cross3(const float a[3], const float b[3], float o[3]) {
    o[0] = a[1] * b[2] - a[2] * b[1];
    o[1] = a[2] * b[0] - a[0] * b[2];
    o[2] = a[0] * b[1] - a[1] * b[0];
}
__device__ __forceinline__ float dot3(const float a[3], const float b[3]) {
    return a[0] * b[0] + a[1] * b[1] + a[2] * b[2];
}
// select among 4 values by runtime idx (v_cndmask chain, no scratch, no EXEC change)
__device__ __forceinline__ float sel4(float v0, float v1, float v2, float v3, int idx) {
    float r = v0;
    r = (idx == 1) ? v1 : r;
    r = (idx == 2) ? v2 : r;
    r = (idx == 3) ? v3 : r;
    return r;
}
// constant-lane broadcast via v_readlane_b32 -> SGPR (no LDS round-trip, no DScnt)
__device__ __forceinline__ float readlane_f(float v, int srcLane) {
    return __int_as_float(__builtin_amdgcn_readlane(__float_as_int(v), srcLane));
}

__device__ void dihedral(const float r12[3], const float r23[3], const float r34[3],
                         float* alpha, float grad[4][3]) {
    float A[3], B[3], C[3];
    cross3(r12, r23, A);
    cross3(r23, r34, B);
    float cos_alpha = dot3(A, B);
    float n23 = sqrtf(dot3(r23, r23));
    float sin_alpha = dot3(A, r34) * n23;
    *alpha = atan2f(sin_alpha, cos_alpha) * DEGF;
    float rA = sqrtf(dot3(A, A));
    float rB = sqrtf(dot3(B, B));
    cross3(r23, A, C);
    float rC = sqrtf(dot3(C, C));
    float cos_phi = dot3(A, B) / (rA * rB);
    float sin_phi = dot3(C, B) / (rC * rB);
    float Bn[3] = {B[0] / rB, B[1] / rB, B[2] / rB};
    float f0[3], f1[3], f2[3];
    if (fabsf(sin_phi) > 0.1f) {  // wave-uniform branch
        float An[3] = {A[0] / rA, A[1] / rA, A[2] / rA};
        float dcA[3], dcB[3];
        #pragma unroll
        for (int i = 0; i < 3; ++i) {
            dcA[i] = (cos_phi * An[i] - Bn[i]) / rA;
            dcB[i] = (cos_phi * Bn[i] - An[i]) / rB;
        }
        float K = (1.0f / sin_phi) * DEGF;
        float t0[3], t2[3], ta[3], tb[3];
        cross3(r23, dcA, t0);
        cross3(dcB, r23, t2);
        cross3(dcA, r12, ta);
        cross3(r34, dcB, tb);
        #pragma unroll
        for (int i = 0; i < 3; ++i) {
            f0[i] = K * t0[i];
            f2[i] = K * t2[i];
            f1[i] = K * (ta[i] + tb[i]);
        }
    } else {
        float Cn[3] = {C[0] / rC, C[1] / rC, C[2] / rC};
        float dC[3], dB[3];
        #pragma unroll
        for (int i = 0; i < 3; ++i) {
            dC[i] = (sin_phi * Cn[i] - Bn[i]) / rC;
            dB[i] = (sin_phi * Bn[i] - Cn[i]) / rB;
        }
        float K = -(1.0f / cos_phi) * DEGF;
        f0[0] = K * ((r23[1] * r23[1] + r23[2] * r23[2]) * dC[0] - r23[0] * r23[1] * dC[1] - r23[0] * r23[2] * dC[2]);
        f0[1] = K * ((r23[2] * r23[2] + r23[0] * r23[0]) * dC[1] - r23[1] * r23[2] * dC[2] - r23[1] * r23[0] * dC[0]);
        f0[2] = K * ((r23[0] * r23[0] + r23[1] * r23[1]) * dC[2] - r23[2] * r23[0] * dC[0] - r23[2] * r23[1] * dC[1]);
        float t2[3];
        cross3(dB, r23, t2);
        #pragma unroll
        for (int i = 0; i < 3; ++i) f2[i] = K * t2[i];
        f1[0] = K * (-(r23[1] * r12[1] + r23[2] * r12[2]) * dC[0] + (2.f * r23[0] * r12[1] - r12[0] * r23[1]) * dC[1] +
                     (2.f * r23[0] * r12[2] - r12[0] * r23[2]) * dC[2] + dB[2] * r34[1] - dB[1] * r34[2]);
        f1[1] = K * (-(r23[2] * r12[2] + r23[0] * r12[0]) * dC[1] + (2.f * r23[1] * r12[2] - r12[1] * r23[2]) * dC[2] +
                     (2.f * r23[1] * r12[0] - r12[1] * r23[0]) * dC[0] + dB[0] * r34[2] - dB[2] * r34[0]);
        f1[2] = K * (-(r23[0] * r12[0] + r23[1] * r12[1]) * dC[2] + (2.f * r23[2] * r12[0] - r12[2] * r23[0]) * dC[0] +
                     (2.f * r23[2] * r12[1] - r12[2] * r23[1]) * dC[1] + dB[1] * r34[0] - dB[0] * r34[1]);
    }
    #pragma unroll
    for (int i = 0; i < 3; ++i) {
        grad[0][i] = -f0[i];
        grad[1][i] = -f1[i] + f0[i];
        grad[2][i] = -f2[i] + f1[i];
        grad[3][i] = f2[i];
    }
}

__global__ __launch_bounds__(32) void alad_metad_kernel(
    const float* __restrict__ position, const float* __restrict__ lattice,
    const float* __restrict__ metad_energy, const float* __restrict__ metad_grad,
    const int* __restrict__ step_p, float* __restrict__ out) {
    const int lane = threadIdx.x;
    const int row = lane & 15;
    const bool lo = lane < 16;

    // ---- uniform loads (SMEM path; same in every lane) ----
    float pos[3][5];
    #pragma unroll
    for (int r = 0; r < 3; ++r)
        #pragma unroll
        for (int c = 0; c < 5; ++c) pos[r][c] = position[r * 5 + c];
    float U[3][3];
    #pragma unroll
    for (int i = 0; i < 3; ++i)
        #pragma unroll
        for (int j = 0; j < 3; ++j) U[i][j] = lattice[i * 3 + j];

    // bond vectors: vec[k] = atom[k+1] - atom[k]
    float vec[4][3];
    #pragma unroll
    for (int k = 0; k < 4; ++k)
        #pragma unroll
        for (int d = 0; d < 3; ++d) vec[k][d] = pos[d][k + 1] - pos[d][k];

    // reciprocal lattice rows
    float recip[3][3];
    #pragma unroll
    for (int i = 0; i < 3; ++i) {
        const int i1 = (i + 1) % 3, i2 = (i + 2) % 3;
        float v[3];
        cross3(U[i1], U[i2], v);
        float den = dot3(v, U[i]);
        #pragma unroll
        for (int j = 0; j < 3; ++j) recip[i][j] = v[j] / den;
    }

    // ---- WMMA #1: S(4x3) = vec(4x3, K-padded to 4) x recip^T(3x3) ----
    // A 16x4 f32 fragment: lanes 0-15 -> (M=lane, K0/K1), lanes 16-31 -> (M=lane-16, K2/K3)
    float vA0 = sel4(vec[0][0], vec[1][0], vec[2][0], vec[3][0], row);
    float vA1 = sel4(vec[0][1], vec[1][1], vec[2][1], vec[3][1], row);
    float vA2 = sel4(vec[0][2], vec[1][2], vec[2][2], vec[3][2], row);
    v2f a1;
    a1.x = (row < 4) ? (lo ? vA0 : vA2) : 0.0f;
    a1.y = (row < 4 && lo) ? vA1 : 0.0f;
    // B 4x16 fragment: B[k][n] = recip[n][k]; lanes 0-15 -> K0/K1, lanes 16-31 -> K2/K3, N = lane&15
    float rB0 = sel4(recip[0][0], recip[1][0], recip[2][0], 0.0f, row);
    float rB1 = sel4(recip[0][1], recip[1][1], recip[2][1], 0.0f, row);
    float rB2 = sel4(recip[0][2], recip[1][2], recip[2][2], 0.0f, row);
    v2f b1;
    b1.x = (row < 3) ? (lo ? rB0 : rB2) : 0.0f;
    b1.y = (row < 3 && lo) ? rB1 : 0.0f;
    v8f cz = {};
    v8f d1 = __builtin_amdgcn_wmma_f32_16x16x4_f32(false, a1, false, b1, (short)0, cz, false, false);

    // broadcast S[m][n] (VGPR m, lane n) to all lanes via v_readlane, round to shift count
    float sr[4][3];
    #pragma unroll
    for (int m = 0; m < 4; ++m)
        #pragma unroll
        for (int n = 0; n < 3; ++n) sr[m][n] = floorf(readlane_f(d1[m], n) + 0.5f);

    // ---- WMMA #2: wrapped(4x3) = (-floor_S)(4x3) x U(3x3) + vec(4x3)  [A*B + C] ----
    float sA0 = sel4(sr[0][0], sr[1][0], sr[2][0], sr[3][0], row);
    float sA1 = sel4(sr[0][1], sr[1][1], sr[2][1], sr[3][1], row);
    float sA2 = sel4(sr[0][2], sr[1][2], sr[2][2], sr[3][2], row);
    v2f a2;
    a2.x = (row < 4) ? (lo ? -sA0 : -sA2) : 0.0f;
    a2.y = (row < 4 && lo) ? -sA1 : 0.0f;
    float u0 = sel4(U[0][0], U[0][1], U[0][2], 0.0f, row);
    float u1 = sel4(U[1][0], U[1][1], U[1][2], 0.0f, row);
    float u2 = sel4(U[2][0], U[2][1], U[2][2], 0.0f, row);
    v2f b2;
    b2.x = (row < 3) ? (lo ? u0 : u2) : 0.0f;
    b2.y = (row < 3 && lo) ? u1 : 0.0f;
    // C 16x16 f32: element (M,N) at VGPR M, lane N (M<8); place vec[m][n] at m<4, n<3
    v8f c2 = {};
    #pragma unroll
    for (int m = 0; m < 4; ++m)
        c2[m] = lo ? sel4(vec[m][0], vec[m][1], vec[m][2],
                          0.0f, (row < 3) ? row : 3)
                   : 0.0f;
    v8f d2 = __builtin_amdgcn_wmma_f32_16x16x4_f32(false, a2, false, b2, (short)0, c2, false, false);

    float w[4][3];  // r12, r23, r34, r45 (wrapped), identical in all lanes after readlane
    #pragma unroll
    for (int m = 0; m < 4; ++m)
        #pragma unroll
        for (int n = 0; n < 3; ++n) w[m][n] = readlane_f(d2[m], n);

    // ---- dihedrals (uniform VALU) ----
    float phi, psi, pg[4][3], sg[4][3];
    dihedral(w[0], w[1], w[2], &phi, pg);
    dihedral(w[1], w[2], w[3], &psi, sg);

    int i_phi = (int)floorf((phi + 180.0f) * 0.2f);
    i_phi = i_phi < 0 ? 0 : (i_phi > 71 ? 71 : i_phi);
    int i_psi = (int)floorf((psi + 180.0f) * 0.2f);
    i_psi = i_psi < 0 ? 0 : (i_psi > 71 ? 71 : i_psi);

    const int cell = i_phi * 72 + i_psi;
    float g0 = metad_grad[cell * 2 + 0];
    float g1 = metad_grad[cell * 2 + 1];

    const int step = step_p[0];
    if (step > 0 && (step % 1000) == 0) {  // wave-uniform
        // hill deposition collapses to the single consumed cell
        float wt = expf(-metad_energy[cell] * (1.0f / (0.0019872041f * 3000.0f)));
        float cphi = -177.5f + 5.0f * (float)i_phi;
        float cpsi = -177.5f + 5.0f * (float)i_psi;
        float dphi = phi - cphi;
        dphi = (dphi > 180.0f) ? dphi - 360.0f : ((dphi < -180.0f) ? dphi + 360.0f : dphi);
        float dpsi = psi - cpsi;
        dpsi = (dpsi > 180.0f) ? dpsi - 360.0f : ((dpsi < -180.0f) ? dpsi + 360.0f : dpsi);
        const float inv2sig2 = 1.0f / 450.0f;  // 2*SIG2 = 450
        float t2p = dphi * inv2sig2;
        float t2s = dpsi * inv2sig2;
        float he = wt * expf(-(t2p * dphi + t2s * dpsi));
        g0 += 2.0f * he * t2p;
        g1 += 2.0f * he * t2s;
    }
    g0 = -g0;
    g1 = -g1;

    // applied_force[r][c] = g0*phi_grad[c][r] (c<4) + g1*psi_grad[c-1][r] (c>=1)
    if (lane == 0) {
        #pragma unroll
        for (int r = 0; r < 3; ++r)
            #pragma unroll
            for (int c = 0; c < 5; ++c) {
                float val = 0.0f;
                if (c < 4) val += g0 * pg[c][r];
                if (c >= 1) val += g1 * sg[c - 1][r];
                out[r * 5 + c] = val;
            }
    }
}

extern "C" void kernel_launch(void* const* d_in, const int* in_sizes, int n_in,
                              void* d_out, int out_size, void* d_ws, size_t ws_size,
                              hipStream_t stream) {
    (void)in_sizes; (void)n_in; (void)out_size; (void)d_ws; (void)ws_size;
    const float* position     = (const float*)d_in[0];
    const float* lattice      = (const float*)d_in[4];
    const float* metad_energy = (const float*)d_in[5];
    const float* metad_grad   = (const float*)d_in[6];
    const int*   step         = (const int*)d_in[7];
    alad_metad_kernel<<<1, 32, 0, stream>>>(position, lattice, metad_energy, metad_grad,
                                            step, (float*)d_out);
}